// CTCLayer_2345052143976
// MI455X (gfx1250) — compile-verified
//
#include <hip/hip_runtime.h>

// CTC batch cost (keras ctc_batch_cost semantics, blank = C-1), B=1024, T=256, C=128, L=48.
// One wave32 per batch element; 4 contiguous extended states per lane (S=97 <= 128).
// Rows of y_pred staged global->LDS with CDNA5 async loads, double buffered.
// All DP math in log2 domain (native v_exp_f32 / v_log_f32 are base-2).

constexpr int T_ = 256;
constexpr int C_ = 128;
constexpr int L_ = 48;

#define NEGV (-1.0e30f)

#if __has_builtin(__builtin_amdgcn_exp2f)
#define EXP2F(x) __builtin_amdgcn_exp2f(x)
#else
#define EXP2F(x) exp2f(x)
#endif
#if __has_builtin(__builtin_amdgcn_logf)
#define LOG2F(x) __builtin_amdgcn_logf(x)   // v_log_f32 = log2
#else
#define LOG2F(x) log2f(x)
#endif

// Async copy of 16 bytes/lane into LDS (gfx1250). vdst = per-lane LDS byte address
// (wave-relative), vaddr = per-lane 64-bit global address. Tracked by ASYNCcnt.
// IMPORTANT: ldsoff must be derived from the __shared__ pointer (ptrtoint) so the
// LDS object escapes and the "memory" clobber is seen as a potential writer of it.
#define ASYNC_ROW_LOAD(ldsoff, gptr)                                      \
  asm volatile("global_load_async_to_lds_b128 %0, %1, off"                \
               :: "v"(ldsoff), "v"(gptr) : "memory")

__device__ __forceinline__ float lse2_2(float a, float b) {
  float m = fmaxf(a, b);
  return m + LOG2F(EXP2F(a - m) + EXP2F(b - m));
}
__device__ __forceinline__ float lse3_2(float a, float b, float c) {
  float m = fmaxf(a, fmaxf(b, c));
  return m + LOG2F(EXP2F(a - m) + EXP2F(b - m) + EXP2F(c - m));
}

__global__ __launch_bounds__(32) void ctc_loss_kernel(
    const float* __restrict__ yp, const int* __restrict__ labels,
    float* __restrict__ out) {
  __shared__ float smem[2 * C_];            // two 512B row buffers

  const int b    = blockIdx.x;
  const int lane = threadIdx.x;
  const float eps = 1e-7f;
  const int blank = C_ - 1;

  // Low 32 bits of the flat pointer to an LDS object == wave-relative LDS byte
  // offset (flat LDS addr = {aperture_hi32, as3_offset_lo32}). The ptrtoint also
  // escapes smem so the asm memory clobbers are treated as potential writers.
  const unsigned lds_base = (unsigned)(uintptr_t)(void*)smem;
  const unsigned lds0 = lds_base + lane * 16;          // buffer 0, 16B per lane
  const unsigned lds1 = lds_base + 512 + lane * 16;    // buffer 1

  const float* __restrict__ grow = yp + (size_t)b * T_ * C_;
  const int*   __restrict__ lb   = labels + (size_t)b * L_;

  // Lane l owns extended states s = 4l..4l+3. Even s -> blank, odd s -> label[(s-1)/2].
  const int j1 = 2 * lane;       // label index for slot1 (s = 4l+1)
  const int j3 = 2 * lane + 1;   // label index for slot3 (s = 4l+3)
  int e1 = blank, e3 = blank;
  int lab1 = -1, lab3 = -1, labm1 = -2;
  if (j1 < L_) { lab1 = lb[j1]; e1 = lab1; }
  if (j3 < L_) { lab3 = lb[j3]; e3 = lab3; }
  if (lane > 0 && (j1 - 1) < L_) labm1 = lb[j1 - 1];
  const bool cs1 = (lane > 0) && (j1 < L_) && (lab1 != labm1);  // skip into s=4l+1
  const bool cs3 = (j3 < L_) && (lab3 != lab1);                 // skip into s=4l+3

  // Prime the pipeline: rows 0 and 1.
  ASYNC_ROW_LOAD(lds0, grow + lane * 4);
  ASYNC_ROW_LOAD(lds1, grow + C_ + lane * 4);

  float a0 = NEGV, a1 = NEGV, a2 = NEGV, a3 = NEGV;

  // t = 0: alpha[0] = lp(blank), alpha[1] = lp(label0), else NEG.
  asm volatile("s_wait_asynccnt 0x1" ::: "memory");   // row 0 landed
  {
    float pb = smem[blank];
    float p1 = smem[e1];
    asm volatile("s_wait_dscnt 0x0" ::: "memory");    // buffer 0 reads done
    ASYNC_ROW_LOAD(lds0, grow + 2 * C_ + lane * 4);   // row 2 -> buffer 0
    if (lane == 0) {
      a0 = LOG2F(pb + eps);
      a1 = LOG2F(p1 + eps);
    }
  }

#pragma unroll 2
  for (int t = 1; t < T_; ++t) {
    const int cur = t & 1;
    if (t < T_ - 1) {
      asm volatile("s_wait_asynccnt 0x1" ::: "memory");  // row t landed (row t+1 may fly)
    } else {
      asm volatile("s_wait_asynccnt 0x0" ::: "memory");  // last row
    }
    const float* buf = smem + cur * C_;
    float qb = buf[blank];
    float q1 = buf[e1];
    float q3 = buf[e3];
    float lpb = LOG2F(qb + eps);
    float lp1 = LOG2F(q1 + eps);
    float lp3 = LOG2F(q3 + eps);
    asm volatile("s_wait_dscnt 0x0" ::: "memory");       // LDS reads of this buffer done
    if (t + 2 < T_) {
      ASYNC_ROW_LOAD(cur ? lds1 : lds0, grow + (t + 2) * C_ + lane * 4);
    }

    // alpha[s-1] for s=4l comes from lane l-1 slot3.
    float u3 = __shfl_up(a3, 1);
    if (lane == 0) u3 = NEGV;

    float n0 = lse2_2(a0, u3) + lpb;                       // even state: no skip
    float n1 = lse3_2(a1, a0, cs1 ? u3 : NEGV) + lp1;      // odd state
    float n2 = lse2_2(a2, a1) + lpb;                       // even state: no skip
    float n3 = lse3_2(a3, a2, cs3 ? a1 : NEGV) + lp3;      // odd state
    a0 = n0; a1 = n1; a2 = n2; a3 = n3;
  }

  // loss = -ln(exp(alpha[95]) + exp(alpha[96])); state 95 = lane23 slot3, 96 = lane24 slot0.
  float a95 = __shfl(a3, 23);
  float a96 = __shfl(a0, 24);
  if (lane == 0) {
    float l2 = lse2_2(a95, a96);
    out[b] = -l2 * 0.69314718055994530942f;  // convert log2 -> ln
  }
}

extern "C" void kernel_launch(void* const* d_in, const int* in_sizes, int n_in,
                              void* d_out, int out_size, void* d_ws, size_t ws_size,
                              hipStream_t stream) {
  const float* y      = (const float*)d_in[0];
  const int*   labels = (const int*)d_in[1];
  float*       out    = (float*)d_out;
  const int B = in_sizes[0] / (T_ * C_);   // 1024
  hipLaunchKernelGGL(ctc_loss_kernel, dim3(B), dim3(32), 0, stream, y, labels, out);
}